// SPDLogEuclideanMean_78915729097347
// MI455X (gfx1250) — compile-verified
//
#include <hip/hip_runtime.h>
#include <math.h>

// ---------------------------------------------------------------------------
// SPD log-Euclidean mean, matmul-only formulation for CDNA5 (gfx1250).
//   logm(X): scale by Frobenius norm -> 5 Newton-Schulz square roots ->
//            8-term Mercator series (all 64x64 matmuls via v_wmma_f32_16x16x4_f32)
//   expm(S): scale by 2^-6 -> 10-term Taylor (Horner) -> 6 squarings
// Deterministic: fixed-order accumulation, no atomics.
// ---------------------------------------------------------------------------

typedef __attribute__((ext_vector_type(2))) float v2f;
typedef __attribute__((ext_vector_type(8))) float v8f;

namespace {
constexpr int DD = 64;                 // matrix dimension
constexpr int NB = 64;                 // batch count
constexpr int NN = 200;                // matrices per batch
constexpr int GG = 25;                 // blocks per batch in phase 1
constexpr int PERG = NN / GG;          // 8 matrices per block
constexpr int ST = 66;                 // LDS row stride (bank-conflict padding)
constexpr int MAT = DD * DD;           // 4096
constexpr int NS_ITERS = 10;           // Newton-Schulz iterations per sqrt level
constexpr int SQRT_LEVELS = 5;         // inverse scaling-and-squaring depth
constexpr int LOG_TERMS = 8;           // Mercator series terms
constexpr int EXP_TERMS = 10;          // Taylor terms for expm
constexpr int EXP_SQUARES = 6;         // squarings for expm
}

// 64x64 matmul C = alpha * A * B, cooperatively by 8 wave32 waves (256 thr).
// Each wave owns two 16x16 output tiles in the SAME tile column (ni shared),
// so the B fragment is loaded once per K-step and fed to two WMMAs.
// Syncs at ENTRY only: one barrier separates consecutive matmul phases.
__device__ __forceinline__ void mm64(float* __restrict__ C,
                                     const float* __restrict__ A,
                                     const float* __restrict__ Bm,
                                     float alpha) {
  const int tid  = threadIdx.x;
  const int lane = tid & 31;
  const int wave = tid >> 5;
  const int hl   = lane >> 4;           // half-wave K split (ISA A/B layout)
  const int l15  = lane & 15;
  const int mi0  = wave >> 2;           // tile rows mi0 and mi0+2
  const int ni   = wave & 3;            // shared tile column
  const int arow0 = (mi0 * 16 + l15) * ST;
  const int arow1 = arow0 + 32 * ST;
  const int bcol  = ni * 16 + l15;
  __syncthreads();
  v8f c0 = {0.f, 0.f, 0.f, 0.f, 0.f, 0.f, 0.f, 0.f};
  v8f c1 = {0.f, 0.f, 0.f, 0.f, 0.f, 0.f, 0.f, 0.f};
#pragma unroll
  for (int kc = 0; kc < 16; ++kc) {
    const int k0 = kc * 4 + hl * 2;
    v2f a0, a1, bv;
    // A 16x4 fragments: lanes 0-15 carry K={k0,k0+1}, lanes 16-31 K={k0+2,k0+3}
    a0.x = A[arow0 + k0];
    a0.y = A[arow0 + k0 + 1];
    a1.x = A[arow1 + k0];
    a1.y = A[arow1 + k0 + 1];
    // B 4x16 fragment, shared by both output tiles
    bv.x = Bm[k0 * ST + bcol];
    bv.y = Bm[(k0 + 1) * ST + bcol];
    c0 = __builtin_amdgcn_wmma_f32_16x16x4_f32(false, a0, false, bv,
                                               (short)0, c0, false, false);
    c1 = __builtin_amdgcn_wmma_f32_16x16x4_f32(false, a1, false, bv,
                                               (short)0, c1, false, false);
  }
  // D layout: VGPR r -> row base + r + 8*half, col across lanes
  const int r0 = mi0 * 16 + hl * 8;
#pragma unroll
  for (int r = 0; r < 8; ++r)
    C[(r0 + r) * ST + bcol] = alpha * c0[r];
#pragma unroll
  for (int r = 0; r < 8; ++r)
    C[(r0 + 32 + r) * ST + bcol] = alpha * c1[r];
}

// Elementwise helpers: barrier at entry AND exit (cheap; guards WAR vs the
// entry-only-synced matmuls on both sides, incl. pointer-swapped buffers).
__device__ __forceinline__ void diag_add(float* M, float v) {
  __syncthreads();
  const int tid = threadIdx.x;
  if (tid < DD) M[tid * ST + tid] += v;
  __syncthreads();
}

__device__ __forceinline__ void set_ident(float* M, float v, int row, int colb) {
  __syncthreads();
#pragma unroll
  for (int j = 0; j < 16; ++j)
    M[row * ST + colb + j] = (row == colb + j) ? v : 0.f;
  __syncthreads();
}

// Phase 1: each block computes sum_{n in its slice} logm(X[n][b]) -> partial
__global__ __launch_bounds__(256) void spd_logsum_kernel(
    const float* __restrict__ X, float* __restrict__ partial) {
  __shared__ float bufY[DD * ST];
  __shared__ float bufZ[DD * ST];
  __shared__ float bufT[DD * ST];
  __shared__ float bufW[DD * ST];
  __shared__ float red[256];
  __shared__ float scal;

  const int tid  = threadIdx.x;
  const int g    = blockIdx.x;          // group within batch
  const int b    = blockIdx.y;          // batch
  const int row  = tid >> 2;            // owned row (16 contiguous elements)
  const int colb = (tid & 3) * 16;

  const float dco[LOG_TERMS] = {1.f, -0.5f, 1.f / 3.f, -0.25f,
                                0.2f, -1.f / 6.f, 1.f / 7.f, -0.125f};

  float acc[16];
#pragma unroll
  for (int j = 0; j < 16; ++j) acc[j] = 0.f;

  for (int m = 0; m < PERG; ++m) {
    const int n = g * PERG + m;
    const float* Xn = X + ((size_t)n * NB + b) * MAT;
    if (m + 1 < PERG)  // warm L2 for the next matrix while we iterate
      __builtin_prefetch(X + ((size_t)(n + 1) * NB + b) * MAT + tid * 16, 0, 0);

    float* pY = bufY;
    float* pZ = bufZ;
    float* pT = bufT;
    float* pW = bufW;

    // --- load X into Y, Frobenius norm reduction ---
    float fro = 0.f;
    const float4* src = (const float4*)(Xn + row * DD + colb);
#pragma unroll
    for (int q = 0; q < 4; ++q) {
      const float4 v = src[q];
      float* d = pY + row * ST + colb + q * 4;
      d[0] = v.x; d[1] = v.y; d[2] = v.z; d[3] = v.w;
      fro += v.x * v.x + v.y * v.y + v.z * v.z + v.w * v.w;
    }
    red[tid] = fro;
    __syncthreads();
    for (int s = 128; s > 0; s >>= 1) {
      if (tid < s) red[tid] += red[tid + s];
      __syncthreads();
    }
    if (tid == 0) scal = sqrtf(red[0]) + 1e-30f;
    __syncthreads();
    const float c = scal;
    const float rc = 1.0f / c;
    const float lnc = logf(c);

    // Y = X / c  (spectrum in (0,1]);  Z = I
#pragma unroll
    for (int j = 0; j < 16; ++j) {
      pY[row * ST + colb + j] *= rc;
      pZ[row * ST + colb + j] = (row == colb + j) ? 1.f : 0.f;
    }
    __syncthreads();

    // --- SQRT_LEVELS repeated square roots via coupled Newton-Schulz ---
    for (int lev = 0; lev < SQRT_LEVELS; ++lev) {
      for (int it = 0; it < NS_ITERS; ++it) {
        // T = 1.5 I - 0.5 Z Y
        mm64(pT, pZ, pY, -0.5f);
        diag_add(pT, 1.5f);
        if (it == NS_ITERS - 1) {
          // Z is reset for the next level: skip the dead newZ product
          mm64(pZ, pY, pT, 1.0f);                  // newY -> old Z buffer
          float* t0 = pY; pY = pZ; pZ = t0;
        } else {
          mm64(pW, pT, pZ, 1.0f);                  // newZ
          mm64(pZ, pY, pT, 1.0f);                  // newY -> old Z buffer
          float* t0 = pY; pY = pZ; pZ = pW; pW = t0;
        }
      }
      if (lev + 1 < SQRT_LEVELS)
        set_ident(pZ, 1.f, row, colb);             // Y holds sqrt; Z = I
    }

    // --- A = R - I ; log(I+A) = A * p(A) via Horner ---
    diag_add(pY, -1.f);
    set_ident(pZ, dco[LOG_TERMS - 1], row, colb);
#pragma unroll
    for (int j = LOG_TERMS - 2; j >= 0; --j) {
      mm64(pW, pY, pZ, 1.0f);
      diag_add(pW, dco[j]);
      float* t0 = pZ; pZ = pW; pW = t0;
    }
    mm64(pW, pY, pZ, 1.0f);  // L = A * p(A)

    // logm(X) = 2^SQRT_LEVELS * L + ln(c) I ; accumulate (fixed order)
    __syncthreads();
    const float s2k = (float)(1 << SQRT_LEVELS);
#pragma unroll
    for (int j = 0; j < 16; ++j)
      acc[j] += s2k * pW[row * ST + colb + j] +
                ((row == colb + j) ? lnc : 0.f);
    __syncthreads();
  }

  float* dst = partial + ((size_t)b * GG + g) * MAT + tid * 16;
#pragma unroll
  for (int j = 0; j < 16; ++j) dst[j] = acc[j];
}

// Phase 2: reduce partials, S = mean log, out = expm(S)
__global__ __launch_bounds__(256) void spd_expm_kernel(
    const float* __restrict__ partial, float* __restrict__ out) {
  __shared__ float bufS[DD * ST];
  __shared__ float bufH[DD * ST];
  __shared__ float bufW[DD * ST];

  const int tid  = threadIdx.x;
  const int b    = blockIdx.x;
  const int row  = tid >> 2;
  const int colb = (tid & 3) * 16;

  const float rf[EXP_TERMS + 1] = {
      1.f, 1.f, 0.5f, 1.f / 6.f, 1.f / 24.f, 1.f / 120.f,
      1.f / 720.f, 1.f / 5040.f, 1.f / 40320.f, 1.f / 362880.f,
      1.f / 3628800.f};

  float s[16];
#pragma unroll
  for (int j = 0; j < 16; ++j) s[j] = 0.f;
  for (int g = 0; g < GG; ++g) {
    const float* p = partial + ((size_t)b * GG + g) * MAT + tid * 16;
#pragma unroll
    for (int j = 0; j < 16; ++j) s[j] += p[j];
  }

  // S_scaled = (sum / N) * 2^-EXP_SQUARES
  const float sc = (1.0f / (float)NN) * (1.0f / (float)(1 << EXP_SQUARES));
#pragma unroll
  for (int j = 0; j < 16; ++j) bufS[row * ST + colb + j] = s[j] * sc;
  // H = rf[EXP_TERMS] * I
#pragma unroll
  for (int j = 0; j < 16; ++j)
    bufH[row * ST + colb + j] = (row == colb + j) ? rf[EXP_TERMS] : 0.f;
  __syncthreads();

  float* pH = bufH;
  float* pW = bufW;
#pragma unroll
  for (int j = EXP_TERMS - 1; j >= 0; --j) {
    mm64(pW, bufS, pH, 1.0f);
    diag_add(pW, rf[j]);
    float* t0 = pH; pH = pW; pW = t0;
  }
  // pH = exp(S/2^m); square m times
  for (int q = 0; q < EXP_SQUARES; ++q) {
    mm64(pW, pH, pH, 1.0f);
    float* t0 = pH; pH = pW; pW = t0;
  }

  __syncthreads();
  float* dst = out + (size_t)b * MAT + tid * 16;
#pragma unroll
  for (int j = 0; j < 16; ++j) dst[j] = pH[row * ST + colb + j];
}

extern "C" void kernel_launch(void* const* d_in, const int* in_sizes, int n_in,
                              void* d_out, int out_size, void* d_ws,
                              size_t ws_size, hipStream_t stream) {
  (void)in_sizes; (void)n_in; (void)out_size; (void)ws_size;
  const float* X = (const float*)d_in[0];
  float* out = (float*)d_out;
  float* partial = (float*)d_ws;  // NB * GG * 4096 floats = ~26 MB

  dim3 grid1(GG, NB);
  spd_logsum_kernel<<<grid1, 256, 0, stream>>>(X, partial);
  spd_expm_kernel<<<NB, 256, 0, stream>>>(partial, out);
}